// SpeechFrameJudgementLayer_52166672778112
// MI455X (gfx1250) — compile-verified
//
#include <hip/hip_runtime.h>

typedef __attribute__((ext_vector_type(2))) float v2f;
typedef __attribute__((ext_vector_type(8))) float v8f;

// One wave32 block. Lane i (i<6) evaluates combo i's two interval indicators,
// a single V_WMMA_F32_16X16X4_F32 forms the outer product whose diagonal is
// the per-combo AND, and a wave32 ballot gives any()/first-true in one op.
__global__ void __launch_bounds__(32)
speech_judge_kernel(const float* __restrict__ score, float* __restrict__ out) {
    const int lane = threadIdx.x;

    // Combo tables (condition 0 is always on score[0]).
    const float lo0[6] = {0.6f, 0.5f, 0.5f, 0.5f, 0.5f, 0.5f};
    const float hi0[6] = {1.0f, 0.7f, 0.7f, 0.7f, 0.7f, 0.7f};
    const int   c1 [6] = {0, 2, 3, 5, 132, 36};
    const float lo1[6] = {0.0f, 0.3f, 0.2f, 0.2f, 0.2f, 0.1f};
    const float hi1[6] = {1.0f, 0.7f, 0.5f, 0.4f, 0.5f, 0.3f};
    const float sval[6] = {5.0f, 1.0f, 1.0f, 1.5f, 1.0f, 1.2f};

    // Per-lane interval indicators (lanes >= 6 contribute zeros).
    float a0 = 0.0f;  // A[m][0] = t0[m]
    float b0 = 0.0f;  // B[0][n] = t1[n]
    if (lane < 6) {
        const float v0 = score[0];
        a0 = (v0 >= lo0[lane] && v0 <= hi0[lane]) ? 1.0f : 0.0f;
        const float v1 = score[c1[lane]];
        b0 = (v1 >= lo1[lane] && v1 <= hi1[lane]) ? 1.0f : 0.0f;
    }

    // 16x4 f32 A: VGPR0 = K=0 for lanes 0-15 (K=2 in lanes 16-31 -> zero).
    // 4x16 f32 B: VGPR0 = K=0 row across lanes 0-15 -> zero elsewhere.
    v2f a; a[0] = a0; a[1] = 0.0f;
    v2f b; b[0] = b0; b[1] = 0.0f;
    v8f c = {};
    // D = A x B + 0 ; D[i][i] = t0[i]*t1[i] lives in VGPR i of lane i (i<8).
    c = __builtin_amdgcn_wmma_f32_16x16x4_f32(
        /*neg_a=*/false, a, /*neg_b=*/false, b,
        /*c_mod=*/(short)0, c, /*reuse_a=*/false, /*reuse_b=*/false);

    // Lane i (i<6) extracts D[i][i] with constant indices.
    float okv = 0.0f;
    if (lane == 0) okv = c[0];
    if (lane == 1) okv = c[1];
    if (lane == 2) okv = c[2];
    if (lane == 3) okv = c[3];
    if (lane == 4) okv = c[4];
    if (lane == 5) okv = c[5];

    const bool ok = (lane < 6) && (okv > 0.5f);
    const unsigned mask = __builtin_amdgcn_ballot_w32(ok);

    if (lane == 0) {
        if (mask != 0u) {
            const int first = __builtin_ctz(mask);  // argmax of boolean vector
            out[0] = 1.0f;                           // judgement = True
            out[1] = sval[first];                    // already exact at 2 decimals
        } else {
            out[0] = 0.0f;
            out[1] = 0.0f;
        }
    }
}

extern "C" void kernel_launch(void* const* d_in, const int* in_sizes, int n_in,
                              void* d_out, int out_size, void* d_ws, size_t ws_size,
                              hipStream_t stream) {
    (void)in_sizes; (void)n_in; (void)d_ws; (void)ws_size; (void)out_size;
    const float* score = (const float*)d_in[0];
    float* out = (float*)d_out;
    speech_judge_kernel<<<1, 32, 0, stream>>>(score, out);
}